// RepresentationCollapseLoss_57337813402173
// MI455X (gfx1250) — compile-verified
//
#include <hip/hip_runtime.h>
#include <hip/hip_bf16.h>
#include <math.h>

typedef __attribute__((ext_vector_type(2))) float v2f;
typedef __attribute__((ext_vector_type(8))) float v8f;

#define N_ROWS 2048
#define K_DIM  512
#define P_COLS 10
#define P_PAD  16
#define B_ELEMS (K_DIM * P_PAD)   // 8192 floats = 32 KB

// ---------------------------------------------------------------------------
// Init: repack projections (512 x 10 row-major) into padded pair layout
//   Bp[(k>>1)*32 + n*2 + (k&1)] = (n<10) ? proj[k*10+n] : 0
// (so a WMMA B fragment {B[k][n], B[k+1][n]} is one contiguous 8-byte load),
// and fill a ones[] row-scale buffer for the unscaled (z) GEMM.
// ---------------------------------------------------------------------------
__global__ void init_kernel(const float* __restrict__ proj,
                            float* __restrict__ Bp,
                            float* __restrict__ ones) {
    int idx = blockIdx.x * blockDim.x + threadIdx.x;
    if (idx < B_ELEMS) {
        int r = idx & 31;
        int n = r >> 1;
        int k = (idx >> 5) * 2 + (r & 1);
        Bp[idx] = (n < P_COLS) ? proj[k * P_COLS + n] : 0.0f;
    }
    int j = idx - B_ELEMS;
    if (j >= 0 && j < N_ROWS) ones[j] = 1.0f;
}

// ---------------------------------------------------------------------------
// invnorm[row] = 1 / max(||gmm_noise[row]||, 1e-12)
// ---------------------------------------------------------------------------
__global__ __launch_bounds__(128) void rownorm_kernel(const float* __restrict__ gmm,
                                                      float* __restrict__ invnorm) {
    const int row = blockIdx.x;
    const int tid = threadIdx.x;
    float s = 0.0f;
    for (int k = tid; k < K_DIM; k += 128) {
        float v = gmm[row * K_DIM + k];
        s += v * v;
    }
    __shared__ float sd[128];
    sd[tid] = s;
    __syncthreads();
    for (int st = 64; st > 0; st >>= 1) {
        if (tid < st) sd[tid] += sd[tid + st];
        __syncthreads();
    }
    if (tid == 0) invnorm[row] = 1.0f / fmaxf(sqrtf(sd[0]), 1e-12f);
}

// ---------------------------------------------------------------------------
// GEMM: 8 waves / block, one 16-row tile per wave, V_WMMA_F32_16X16X4_F32.
// B (32 KB) is staged into LDS once per block via the CDNA5 async-copy path
// (global_load_async_to_lds_b128, ASYNCcnt), then read back as ds_load_b64.
//   A frag (16x4 f32): lanes 0-15 rows, VGPR0/1 = K{0,1}; lanes 16-31 = K{2,3}
//   B frag (4x16 f32): VGPR0 = K=0 (lanes 0-15) / K=2 (lanes 16-31); VGPR1 = K=1 / K=3
//   C/D (16x16 f32):   VGPR i -> M=i (lanes 0-15), M=i+8 (lanes 16-31), N=lane%16
// blockIdx.y: 0 -> z->post (rowscale = ones) ; 1 -> gmm->prior (rowscale = invnorm)
// ---------------------------------------------------------------------------
__global__ __launch_bounds__(256) void gemm_wmma_kernel(const float* __restrict__ Z,
                                                        const float* __restrict__ G,
                                                        const float* __restrict__ Bp,
                                                        const float* __restrict__ invnorm,
                                                        const float* __restrict__ ones,
                                                        float* __restrict__ post,
                                                        float* __restrict__ prior) {
    __shared__ float sB[B_ELEMS];   // 32 KB

    const int tid   = threadIdx.x;
    const int which = blockIdx.y;                 // 0: z, 1: gmm
    const float* A        = which ? G : Z;
    const float* rowscale = which ? invnorm : ones;
    float* Out            = which ? prior : post;

    // ---- async-stage B into LDS: 256 lanes x 8 iters x 16B = 32 KB --------
    {
#pragma unroll
        for (int it = 0; it < 8; ++it) {
            const int fidx = (it * 256 + tid) * 4;                  // float index
            const unsigned lds_off = (unsigned)(uintptr_t)(&sB[fidx]);
            const unsigned goff    = (unsigned)(fidx * 4);          // byte offset
            asm volatile("global_load_async_to_lds_b128 %0, %1, %2"
                         :
                         : "v"(lds_off), "v"(goff), "s"(Bp)
                         : "memory");
        }
        asm volatile("s_wait_asynccnt 0x0" ::: "memory");
    }
    __syncthreads();

    const int wave = tid >> 5;                    // 0..7
    const int tile = blockIdx.x * 8 + wave;       // 0..127
    const int lane = tid & 31;
    const int half = lane >> 4;                   // 0: K{0,1}, 1: K{2,3}
    const int lrow = lane & 15;

    const float* Arow = A + (tile * 16 + lrow) * K_DIM;

    v8f c = {};
#pragma unroll 8
    for (int k0 = 0; k0 < K_DIM; k0 += 4) {
        const int kk = k0 + 2 * half;
        v2f a;
        a.x = Arow[kk];
        a.y = Arow[kk + 1];
        const float* bptr = &sB[(kk >> 1) * 32 + lrow * 2];
        v2f b;
        b.x = bptr[0];
        b.y = bptr[1];
        c = __builtin_amdgcn_wmma_f32_16x16x4_f32(
                /*neg_a=*/false, a, /*neg_b=*/false, b,
                /*c_mod=*/(short)0, c, /*reuse_a=*/false, /*reuse_b=*/false);
    }

#pragma unroll
    for (int i = 0; i < 8; ++i) {
        const int row = tile * 16 + i + 8 * half;
        Out[row * P_PAD + lrow] = c[i] * rowscale[row];
    }
}

// ---------------------------------------------------------------------------
// Bitonic sort of one 2048-element column in LDS. 20 blocks: 10 cols x {post, prior}.
// ---------------------------------------------------------------------------
__global__ __launch_bounds__(1024) void sort_col_kernel(float* __restrict__ post,
                                                        float* __restrict__ prior) {
    __shared__ float s[N_ROWS];
    const int which = blockIdx.x / P_COLS;
    const int col   = blockIdx.x % P_COLS;
    float* M = which ? prior : post;
    const int tid = threadIdx.x;

    for (int i = tid; i < N_ROWS; i += 1024) s[i] = M[i * P_PAD + col];
    __syncthreads();

    for (int k = 2; k <= N_ROWS; k <<= 1) {
        for (int j = k >> 1; j > 0; j >>= 1) {
            for (int i = tid; i < N_ROWS; i += 1024) {
                const int ixj = i ^ j;
                if (ixj > i) {
                    const float x = s[i];
                    const float y = s[ixj];
                    const bool up = ((i & k) == 0);
                    if (up ? (x > y) : (x < y)) { s[i] = y; s[ixj] = x; }
                }
            }
            __syncthreads();
        }
    }
    for (int i = tid; i < N_ROWS; i += 1024) M[i * P_PAD + col] = s[i];
}

// ---------------------------------------------------------------------------
// Per-column O(N) moment reduction (f64) replacing the O(N^2) pairwise tensor.
//   Su2 = 2N*S4a - 8*S3a*S1a + 6*S2a^2          (same for Sv2 with b)
//   Suv = 2N*Sa2b2 - 4*Sa2b*S1b - 4*Sab2*S1a + 4*Sab^2 + 2*S2a*S2b
//   gw  = (Su2 - 2*Suv + Sv2)/N^2 ;  w = S2a - 2*Sab + S2b
// Reversed-prior branch: same formulas, cross moments paired with b[N-1-i].
// ---------------------------------------------------------------------------
__global__ __launch_bounds__(256) void column_moments_kernel(const float* __restrict__ post,
                                                             const float* __restrict__ prior,
                                                             float* __restrict__ fgw) {
    const int col = blockIdx.x;
    const int tid = threadIdx.x;

    double m[16] = {};
    for (int i = tid; i < N_ROWS; i += 256) {
        const double a  = (double)post[i * P_PAD + col];
        const double b  = (double)prior[i * P_PAD + col];
        const double br = (double)prior[(N_ROWS - 1 - i) * P_PAD + col];
        const double a2 = a * a, b2 = b * b, br2 = br * br;
        m[0]  += a;        m[1]  += a2;       m[2]  += a2 * a;   m[3]  += a2 * a2;
        m[4]  += b;        m[5]  += b2;       m[6]  += b2 * b;   m[7]  += b2 * b2;
        m[8]  += a * b;    m[9]  += a2 * b;   m[10] += a * b2;   m[11] += a2 * b2;
        m[12] += a * br;   m[13] += a2 * br;  m[14] += a * br2;  m[15] += a2 * br2;
    }

    __shared__ double sd[256];
    __shared__ double tot[16];
    for (int q = 0; q < 16; ++q) {
        sd[tid] = m[q];
        __syncthreads();
        for (int st = 128; st > 0; st >>= 1) {
            if (tid < st) sd[tid] += sd[tid + st];
            __syncthreads();
        }
        if (tid == 0) tot[q] = sd[0];
        __syncthreads();
    }

    if (tid == 0) {
        const double N = (double)N_ROWS;
        const double S1a = tot[0], S2a = tot[1], S3a = tot[2], S4a = tot[3];
        const double S1b = tot[4], S2b = tot[5], S3b = tot[6], S4b = tot[7];
        const double sumU2 = 2.0 * N * S4a - 8.0 * S3a * S1a + 6.0 * S2a * S2a;
        const double sumV2 = 2.0 * N * S4b - 8.0 * S3b * S1b + 6.0 * S2b * S2b;

        auto branch = [&](double AB, double A2B, double AB2, double A2B2) {
            const double sumUV = 2.0 * N * A2B2 - 4.0 * A2B * S1b - 4.0 * AB2 * S1a
                               + 4.0 * AB * AB + 2.0 * S2a * S2b;
            const double gw = (sumU2 - 2.0 * sumUV + sumV2) / (N * N);
            const double w  = S2a - 2.0 * AB + S2b;
            return 0.9 * w + 0.1 * gw;   // (1-BETA)*w + BETA*gw
        };
        const double f1 = branch(tot[8],  tot[9],  tot[10], tot[11]);
        const double f2 = branch(tot[12], tot[13], tot[14], tot[15]);
        fgw[col] = (float)fmin(f1, f2);
    }
}

__global__ void finalize_kernel(const float* __restrict__ fgw, float* __restrict__ out) {
    if (blockIdx.x == 0 && threadIdx.x == 0) {
        float s = 0.0f;
        for (int p = 0; p < P_COLS; ++p) s += fgw[p];
        out[0] = s / (float)P_COLS;
    }
}

extern "C" void kernel_launch(void* const* d_in, const int* in_sizes, int n_in,
                              void* d_out, int out_size, void* d_ws, size_t ws_size,
                              hipStream_t stream) {
    const float* z    = (const float*)d_in[0];   // 2048 x 512
    const float* gmm  = (const float*)d_in[1];   // 2048 x 512
    const float* proj = (const float*)d_in[2];   // 512 x 10

    float* ws    = (float*)d_ws;
    float* Bp    = ws;                               // 8192 floats
    float* invn  = Bp + B_ELEMS;                     // 2048 floats
    float* ones  = invn + N_ROWS;                    // 2048 floats
    float* post  = ones + N_ROWS;                    // 32768 floats
    float* prior = post + N_ROWS * P_PAD;            // 32768 floats
    float* fgw   = prior + N_ROWS * P_PAD;           // 10 floats

    init_kernel<<<(B_ELEMS + N_ROWS + 255) / 256, 256, 0, stream>>>(proj, Bp, ones);
    rownorm_kernel<<<N_ROWS, 128, 0, stream>>>(gmm, invn);
    gemm_wmma_kernel<<<dim3(N_ROWS / 16 / 8, 2), 256, 0, stream>>>(z, gmm, Bp, invn, ones,
                                                                   post, prior);
    sort_col_kernel<<<2 * P_COLS, 1024, 0, stream>>>(post, prior);
    column_moments_kernel<<<P_COLS, 256, 0, stream>>>(post, prior, fgw);
    finalize_kernel<<<1, 32, 0, stream>>>(fgw, (float*)d_out);
}